// L1LossMask_50156628083336
// MI455X (gfx1250) — compile-verified
//
#include <hip/hip_runtime.h>
#include <math.h>

// ---------------------------------------------------------------------------
// Masked L1 gather loss for MI455X (gfx1250, wave32).
// Memory-bound gather-reduce; WMMA f32 16x16x4 (ones-matrix) used as the
// in-wave reduction tree. Deterministic two-pass reduction via d_ws.
// ---------------------------------------------------------------------------

typedef float v2f __attribute__((ext_vector_type(2)));
typedef float v8f __attribute__((ext_vector_type(8)));

#define B_      32
#define C_      128
#define HW_     (160 * 160)
#define K_      1024
#define KBLK    256
#define CCHUNK  16
#define NBLK_X  (K_ / KBLK)     // 4
#define NBLK_Y  (C_ / CCHUNK)   // 8
#define NBLOCKS (NBLK_X * NBLK_Y * B_)  // 1024
#define EPS_    1e-5f

// Reduce the 64 values {a0,a1} across all 32 lanes of a wave using the matrix
// unit:  D = A(16x4) * ones(4x16).  A layout (ISA 7.12.2, 32-bit A 16x4):
//   lane l<16 : VGPR0 = A[l,0], VGPR1 = A[l,1]
//   lane l>=16: VGPR0 = A[l-16,2], VGPR1 = A[l-16,3]
// => row-sum R[m] = (lane m a0+a1) + (lane m+16 a0+a1), replicated over N.
// D layout: lane l<16 holds D[0..7, l] in v[0..7]; lane l>=16 holds D[8..15,:].
// Sum v[0..7] per lane, then xor-16 shuffle add => every lane has the total.
__device__ __forceinline__ float wave_reduce_wmma(float a0, float a1) {
    v2f a; a.x = a0;   a.y = a1;
    v2f b; b.x = 1.0f; b.y = 1.0f;   // all-ones B: layout-independent
    v8f c = {};
    v8f d = __builtin_amdgcn_wmma_f32_16x16x4_f32(
        /*neg_a=*/false, a, /*neg_b=*/false, b,
        /*c_mod=*/(short)0, c, /*reuse_a=*/false, /*reuse_b=*/false);
    float s = ((d[0] + d[1]) + (d[2] + d[3])) + ((d[4] + d[5]) + (d[6] + d[7]));
    s += __shfl_xor(s, 16, 32);      // combine M=0..7 half with M=8..15 half
    return s;
}

__global__ __launch_bounds__(KBLK)
void l1mask_partial_kernel(const float* __restrict__ pred,
                           const float* __restrict__ target,
                           const int*   __restrict__ indices,
                           const float* __restrict__ mask,
                           float* __restrict__ ws) {
    const int kb = blockIdx.x;              // k-chunk   0..3
    const int cb = blockIdx.y;              // c-chunk   0..7
    const int b  = blockIdx.z;              // batch     0..31
    const int k  = kb * KBLK + threadIdx.x; // this thread's k

    const int   idx = indices[b * K_ + k];  // spatial gather index
    const float m   = mask[b * K_ + k];

    const float* __restrict__ p =
        pred + (size_t)b * C_ * HW_ + (size_t)(cb * CCHUNK) * HW_ + idx;
    const float* __restrict__ t =
        target + (size_t)b * C_ * K_ + (size_t)(cb * CCHUNK) * K_ + k;

    // 4 independent accumulators; pred loads are the random gather (kept RT so
    // L2 retains reused lines), target is streamed once (non-temporal).
    float acc0 = 0.f, acc1 = 0.f, acc2 = 0.f, acc3 = 0.f;
#pragma unroll
    for (int c = 0; c < CCHUNK; c += 4) {
        const float p0 = p[(size_t)(c + 0) * HW_];
        const float p1 = p[(size_t)(c + 1) * HW_];
        const float p2 = p[(size_t)(c + 2) * HW_];
        const float p3 = p[(size_t)(c + 3) * HW_];
        const float t0 = __builtin_nontemporal_load(&t[(size_t)(c + 0) * K_]);
        const float t1 = __builtin_nontemporal_load(&t[(size_t)(c + 1) * K_]);
        const float t2 = __builtin_nontemporal_load(&t[(size_t)(c + 2) * K_]);
        const float t3 = __builtin_nontemporal_load(&t[(size_t)(c + 3) * K_]);
        acc0 += fabsf(t0 * m - p0 * m);
        acc1 += fabsf(t1 * m - p1 * m);
        acc2 += fabsf(t2 * m - p2 * m);
        acc3 += fabsf(t3 * m - p3 * m);
    }

    // mask sum counted exactly once per (b,k): only by the cb==0 slice
    const float mk = (cb == 0) ? m : 0.0f;

    // wave32 reductions through the matrix unit (uniform control flow here,
    // EXEC is all-ones as WMMA requires)
    const float nsum = wave_reduce_wmma(acc0 + acc2, acc1 + acc3);
    const float msum = wave_reduce_wmma(mk, 0.0f);

    __shared__ float sn[8];
    __shared__ float sm[8];
    const int wid  = threadIdx.x >> 5;
    const int lane = threadIdx.x & 31;
    if (lane == 0) { sn[wid] = nsum; sm[wid] = msum; }
    __syncthreads();

    if (threadIdx.x == 0) {
        float N = 0.f, M = 0.f;
#pragma unroll
        for (int w = 0; w < KBLK / 32; ++w) { N += sn[w]; M += sm[w]; }
        const int bid = (b * NBLK_Y + cb) * NBLK_X + kb;
        ws[bid]           = N;   // numerator partial
        ws[NBLOCKS + bid] = M;   // mask-sum partial (0 for cb>0)
    }
}

__global__ __launch_bounds__(256)
void l1mask_finalize_kernel(const float* __restrict__ ws,
                            float* __restrict__ out) {
    const int t = threadIdx.x;   // 256 threads, single block
    float n = 0.f, m = 0.f;
#pragma unroll
    for (int i = 0; i < NBLOCKS; i += 256) {
        n += ws[i + t];
        m += ws[NBLOCKS + i + t];
    }
    const float nsum = wave_reduce_wmma(n, 0.0f);
    const float msum = wave_reduce_wmma(m, 0.0f);

    __shared__ float sn[8];
    __shared__ float sm[8];
    const int wid  = t >> 5;
    const int lane = t & 31;
    if (lane == 0) { sn[wid] = nsum; sm[wid] = msum; }
    __syncthreads();

    if (t == 0) {
        float N = 0.f, M = 0.f;
#pragma unroll
        for (int w = 0; w < 8; ++w) { N += sn[w]; M += sm[w]; }
        out[0] = N / (M * (float)C_ + EPS_);   // LOSS_WEIGHT == 1.0
    }
}

extern "C" void kernel_launch(void* const* d_in, const int* in_sizes, int n_in,
                              void* d_out, int out_size, void* d_ws, size_t ws_size,
                              hipStream_t stream) {
    const float* pred    = (const float*)d_in[0];   // [B, C, H, W]
    const float* target  = (const float*)d_in[1];   // [B, C, K]
    const int*   indices = (const int*)d_in[2];     // [B, K]
    const float* mask    = (const float*)d_in[3];   // [B, K]
    float*       wsf     = (float*)d_ws;            // 2*NBLOCKS floats (8 KB)
    float*       out     = (float*)d_out;

    (void)in_sizes; (void)n_in; (void)out_size; (void)ws_size;

    dim3 grid(NBLK_X, NBLK_Y, B_);                  // (4, 8, 32) = 1024 blocks
    l1mask_partial_kernel<<<grid, KBLK, 0, stream>>>(pred, target, indices, mask, wsf);
    l1mask_finalize_kernel<<<1, 256, 0, stream>>>(wsf, out);
}